// BoundaryKDV4_24979529793876
// MI455X (gfx1250) — compile-verified
//
#include <hip/hip_runtime.h>
#include <hip/hip_bf16.h>

// ---------------------------------------------------------------------------
// BoundaryKD loss for (B=2, C=14, 96^3) fp32 tensors on MI455X (gfx1250).
// Memory-bound: 297 MB of input -> ~12.7 us floor at 23.3 TB/s.
// Kernel A: single streaming pass  -> pk (f32) + argmax class (u8), 8.8 MB.
// Kernel B: 27-pt class-presence stencil + per-wave chained
//           V_WMMA_F32_16X16X4_F32 accumulation of (num[k], n[k]).
// Kernel C: 64-value finalize -> scalar loss.
// ---------------------------------------------------------------------------

#define NCLS 14
#define HWD  96
#define VOL  (96 * 96 * 96)      // 884736, divisible by 256 and by 32

typedef __attribute__((ext_vector_type(2))) float v2f;
typedef __attribute__((ext_vector_type(8))) float v8f;

// ---------------------------------------------------------------------------
// Kernel A: per-voxel softmax-KL (pk) + argmax class map.
// grid = (VOL/256, B), block = 256. Fully coalesced: lanes walk contiguous v,
// class loop strides by VOL (3.5 MB planes).
// ---------------------------------------------------------------------------
__global__ __launch_bounds__(256) void pk_cls_kernel(
    const float* __restrict__ pS, const float* __restrict__ pT,
    const float* __restrict__ oT, float* __restrict__ pk_out,
    unsigned char* __restrict__ cls_out)
{
    const int v = blockIdx.x * 256 + threadIdx.x;     // < VOL exactly
    const int b = blockIdx.y;
    const size_t base = (size_t)b * NCLS * VOL + (size_t)v;

    float sS[NCLS], sT[NCLS];
    float mS = -3.4e38f, mT = -3.4e38f, mO = -3.4e38f;
    int amax = 0;

#pragma unroll
    for (int c = 0; c < NCLS; ++c) {
        const size_t off = base + (size_t)c * VOL;
        float o = oT[off];
        float a = pS[off];
        float t = pT[off];
        if (c + 4 < NCLS) {   // gfx1250 global_prefetch_b8 on future planes
            __builtin_prefetch(&pS[off + 4 * (size_t)VOL], 0, 0);
            __builtin_prefetch(&pT[off + 4 * (size_t)VOL], 0, 0);
            __builtin_prefetch(&oT[off + 4 * (size_t)VOL], 0, 0);
        }
        sS[c] = a;
        sT[c] = t;
        if (o > mO) { mO = o; amax = c; }   // first-max tie break == argmax
        mS = fmaxf(mS, a);
        mT = fmaxf(mT, t);
    }

    float eS = 0.f, eT = 0.f;
#pragma unroll
    for (int c = 0; c < NCLS; ++c) {
        eS += __expf(sS[c] - mS);
        eT += __expf(sT[c] - mT);
    }
    const float lseS = mS + __logf(eS);
    const float lseT = mT + __logf(eT);

    float pk = 0.f;
#pragma unroll
    for (int c = 0; c < NCLS; ++c) {
        const float lt = sT[c] - lseT;               // log_t
        const float ls = sS[c] - lseS;               // log_s
        pk += __expf(lt) * (lt - ls);                // t * (log_t - log_s)
    }

    pk_out[(size_t)b * VOL + v]  = pk;
    cls_out[(size_t)b * VOL + v] = (unsigned char)amax;
}

// ---------------------------------------------------------------------------
// Kernel B: boundary detection + WMMA reduction.
//   boundary[k,v] = (class k present in 3x3x3 valid neighborhood) AND NOT
//                   (27 valid neighbors all of class k)
//   num[b,k] = sum_v boundary[k,v] * pk[v]   ;   n[b,k] = sum_v boundary[k,v]
// Mapped as Boundary(16x4 classes x voxels) x B(4x16: col0=pk, col1=1)
// chained through V_WMMA_F32_16X16X4_F32 per wave.
// grid = (432, B); 3456 tiles/batch -> 8 uniform iterations per block, so
// EXEC is all-ones at every WMMA (ISA requirement).
// ---------------------------------------------------------------------------
__global__ __launch_bounds__(256) void boundary_wmma_kernel(
    const unsigned char* __restrict__ cls, const float* __restrict__ pk,
    float* __restrict__ gnum, float* __restrict__ gn)
{
    __shared__ float lnum[16];
    __shared__ float ln[16];
    if (threadIdx.x < 16) { lnum[threadIdx.x] = 0.f; ln[threadIdx.x] = 0.f; }
    __syncthreads();

    const int b    = blockIdx.y;
    const int lane = threadIdx.x & 31;
    const int half = lane >> 4;        // 0: K rows {0,1}; 1: K rows {2,3}
    const int m    = lane & 15;        // A-row (class) / B-col (output sel)
    const size_t bbase = (size_t)b * VOL;

    v8f acc = {0.f, 0.f, 0.f, 0.f, 0.f, 0.f, 0.f, 0.f};

    for (int base = blockIdx.x * 256; base < VOL; base += gridDim.x * 256) {
        const int v = base + threadIdx.x;
        const int d = v % HWD;
        const int w = (v / HWD) % HWD;
        const int h = v / (HWD * HWD);

        unsigned mask = 0u;
        int cnt = 0;
#pragma unroll
        for (int dh = -1; dh <= 1; ++dh) {
            const int hh = h + dh;
            if ((unsigned)hh >= (unsigned)HWD) continue;
#pragma unroll
            for (int dw = -1; dw <= 1; ++dw) {
                const int ww = w + dw;
                if ((unsigned)ww >= (unsigned)HWD) continue;
#pragma unroll
                for (int dd = -1; dd <= 1; ++dd) {
                    const int zz = d + dd;
                    if ((unsigned)zz >= (unsigned)HWD) continue;
                    const int c =
                        cls[bbase + (size_t)((hh * HWD + ww) * HWD + zz)];
                    mask |= (1u << c);
                    ++cnt;
                }
            }
        }
        // interior & uniform neighborhood (box==27) -> no boundary classes
        if (cnt == 27 && (mask & (mask - 1u)) == 0u) mask = 0u;

        const float pkv = pk[bbase + v];

        // Per-wave WMMA accumulate: 8 chained 16x16x4 MACs cover 32 voxels.
#pragma unroll
        for (int g = 0; g < 8; ++g) {
            const int k0 = 4 * g + 2 * half;         // this half's K rows
            const unsigned mk0 = __shfl(mask, k0);
            const unsigned mk1 = __shfl(mask, k0 + 1);
            const float    p0  = __shfl(pkv,  k0);
            const float    p1  = __shfl(pkv,  k0 + 1);

            v2f a, bf;
            a.x = (float)((mk0 >> m) & 1u);          // A[m, k0]
            a.y = (float)((mk1 >> m) & 1u);          // A[m, k0+1]
            bf.x = (m == 0) ? p0 : ((m == 1) ? 1.0f : 0.0f);  // B[k0,  m]
            bf.y = (m == 0) ? p1 : ((m == 1) ? 1.0f : 0.0f);  // B[k0+1,m]

            acc = __builtin_amdgcn_wmma_f32_16x16x4_f32(
                false, a, false, bf, (short)0, acc, false, false);
        }
    }

    // D layout: VGPR j, lanes 0-15 -> D[j, lane]; lanes 16-31 -> D[j+8, lane-16]
    // lane 0 : num[0..7]   lane 16: num[8..15]
    // lane 1 : n[0..7]     lane 17: n[8..15]
    if (m <= 1) {
        float* dst = (m == 0) ? lnum : ln;
#pragma unroll
        for (int j = 0; j < 8; ++j) atomicAdd(&dst[half * 8 + j], acc[j]);
    }
    __syncthreads();

    if (threadIdx.x < 16)
        atomicAdd(&gnum[b * 16 + threadIdx.x], lnum[threadIdx.x]);
    else if (threadIdx.x < 32)
        atomicAdd(&gn[b * 16 + (threadIdx.x - 16)], ln[threadIdx.x - 16]);
}

// ---------------------------------------------------------------------------
// Kernel C: loss = sum over (b,k) of num/(C*n) where n>0.
// ---------------------------------------------------------------------------
__global__ void finalize_kernel(const float* __restrict__ gnum,
                                const float* __restrict__ gn,
                                float* __restrict__ out)
{
    __shared__ float s[32];
    const int t = threadIdx.x;            // 32 threads, covers 2*16 entries
    float term = 0.f;
    const float n   = gn[t];
    const float num = gnum[t];
    if (n > 0.f) term = num / ((float)NCLS * n);
    s[t] = term;
    __syncthreads();
    if (t == 0) {
        float accv = 0.f;
        for (int i = 0; i < 32; ++i) accv += s[i];
        out[0] = accv;   // LOSS_WEIGHT * TAU^2 == 1
    }
}

// ---------------------------------------------------------------------------
extern "C" void kernel_launch(void* const* d_in, const int* in_sizes, int n_in,
                              void* d_out, int out_size, void* d_ws,
                              size_t ws_size, hipStream_t stream)
{
    const float* preds_S   = (const float*)d_in[0];
    const float* preds_T   = (const float*)d_in[1];
    const float* outputs_T = (const float*)d_in[2];
    float* out = (float*)d_out;

    const int B = 2;
    char* ws = (char*)d_ws;
    float*         pk   = (float*)ws;                                  // B*VOL f32
    unsigned char* cls  = (unsigned char*)(ws + (size_t)B * VOL * 4);  // B*VOL u8
    float*         gnum = (float*)(ws + (size_t)B * VOL * 4 + (size_t)B * VOL);
    float*         gn   = gnum + B * 16;

    // zero the 64 accumulators (graph-capture-safe memset node)
    hipMemsetAsync(gnum, 0, 2 * B * 16 * sizeof(float), stream);

    pk_cls_kernel<<<dim3(VOL / 256, B), 256, 0, stream>>>(
        preds_S, preds_T, outputs_T, pk, cls);

    boundary_wmma_kernel<<<dim3(432, B), 256, 0, stream>>>(cls, pk, gnum, gn);

    finalize_kernel<<<1, 32, 0, stream>>>(gnum, gn, out);
}